// AmpModule_39865886441859
// MI455X (gfx1250) — compile-verified
//
#include <hip/hip_runtime.h>

#define LX 10
#define LY 10
#define PPHYS 2
#define DD 4
#define CHI 4
#define SLOT 1024   // floats per MPS tensor slot (16*16*4 max)
#define NLANE 32

typedef __attribute__((ext_vector_type(2))) float v2f;
typedef __attribute__((ext_vector_type(8))) float v8f;

static __device__ __forceinline__ int imin(int a, int b) { return a < b ? a : b; }

static __device__ __forceinline__ float wave_sum(float v) {
#pragma unroll
  for (int m = 16; m >= 1; m >>= 1) v += __shfl_xor(v, m, 32);
  return v;
}

static __device__ __forceinline__ void lds_zero(float* p, int n, int lane) {
  for (int t = lane; t < n; t += NLANE) p[t] = 0.0f;
}

// One 16x16 output tile: C = A(16 x Kpad) * B(Kpad x 16).
// Kpad multiple of 4; operands must be zero-padded in K. Whole wave active.
static __device__ __forceinline__ void mm16(const float* A, int lda,
                                            const float* B, int ldb,
                                            float* C, int ldc, int Kpad, int lane) {
  const int half = lane >> 4;
  const int l16  = lane & 15;
  v8f acc = {0.f, 0.f, 0.f, 0.f, 0.f, 0.f, 0.f, 0.f};
  for (int k0 = 0; k0 < Kpad; k0 += 4) {
    const int ka = k0 + (half << 1);  // lanes 0-15: K0/K1, lanes 16-31: K2/K3
    v2f a, b;
    a.x = A[l16 * lda + ka];
    a.y = A[l16 * lda + ka + 1];
    b.x = B[ka * ldb + l16];
    b.y = B[(ka + 1) * ldb + l16];
    acc = __builtin_amdgcn_wmma_f32_16x16x4_f32(false, a, false, b, (short)0, acc,
                                                false, false);
  }
#pragma unroll
  for (int i = 0; i < 8; ++i) C[(half * 8 + i) * ldc + l16] = acc[i];
}

// Gather the selected physical slice of site (i,j) into LDS, applying the
// boundary truncations (first index of each truncated leg).
static __device__ __forceinline__ void load_site(const float* peps, const int* x,
                                                 int i, int j, float* T,
                                                 int& nu, int& nr, int& nd, int& nl,
                                                 int lane) {
  nu = (i == LX - 1) ? 1 : DD;
  nr = (j == LY - 1) ? 1 : DD;
  nd = (i == 0) ? 1 : DD;
  nl = (j == 0) ? 1 : DD;
  const int p = x[i * LY + j];
  const float* src = peps + (((size_t)(i * LY + j) * PPHYS + p) << 8);  // *256
  const int n = nu * nr * nd * nl;
  for (int t = lane; t < n; t += NLANE) {
    int l = t % nl; int q = t / nl;
    int d = q % nd; q /= nd;
    int r = q % nr; int u = q / nr;
    T[t] = src[(u << 6) + (r << 4) + (d << 2) + l];
  }
}

// Canonical MPS compression to bond dim CHI (QR sweep + truncated SVD sweep +
// norm strip).  mps tensors packed (l, r, u=4), slot stride SLOT.
static __device__ void compress_mps(float* mps, int* TL, int* TR,
                                    float* P0, float* P1, float* P2,
                                    float* rowbuf, float* VE,
                                    float* svals, int* perm,
                                    float* logZ, int lane) {
  // ---- left-to-right QR sweep ----
  for (int j = 0; j < LY - 1; ++j) {
    const int l = TL[j], r = TR[j];
    const int m = l * 4;
    const int k = imin(m, r);
    lds_zero(P0, 1024, lane);  // W (work copy of M), 64x16 ld16
    lds_zero(P1, 1024, lane);  // Q, 64x16 ld16
    lds_zero(P2, 1024, lane);  // R, 16x16 ld16 (zero-padded)
    __syncthreads();
    // W[(l*4+u)][c] = mps[j][l][c][u]   (transpose(0,2,1).reshape(l*u, r))
    for (int t = lane; t < m * r; t += NLANE) {
      int c = t % r; int a = t / r;
      int ll = a >> 2, u = a & 3;
      P0[a * 16 + c] = mps[j * SLOT + (ll * r + c) * 4 + u];
    }
    __syncthreads();
    // Modified Gram-Schmidt thin QR (k columns)
    for (int c = 0; c < k; ++c) {
      float part = 0.f;
      for (int t = lane; t < m; t += NLANE) { float w = P0[t * 16 + c]; part += w * w; }
      float nrm = sqrtf(wave_sum(part));
      float inv = (nrm > 1e-30f) ? 1.f / nrm : 0.f;
      if (lane == 0) P2[c * 16 + c] = nrm;
      for (int t = lane; t < m; t += NLANE) P1[t * 16 + c] = P0[t * 16 + c] * inv;
      __syncthreads();
      for (int c2 = c + 1; c2 < r; ++c2) {
        float pd = 0.f;
        for (int t = lane; t < m; t += NLANE) pd += P1[t * 16 + c] * P0[t * 16 + c2];
        float dot = wave_sum(pd);
        if (lane == 0) P2[c * 16 + c2] = dot;
        for (int t = lane; t < m; t += NLANE) P0[t * 16 + c2] -= dot * P1[t * 16 + c];
        __syncthreads();
      }
    }
    __syncthreads();
    // mps[j] = Q reshaped back to (l, k, 4)
    for (int t = lane; t < l * k * 4; t += NLANE) {
      int u = t & 3; int rest = t >> 2;
      int c = rest % k; int ll = rest / k;
      mps[j * SLOT + t] = P1[(ll * 4 + u) * 16 + c];
    }
    // mps[j+1] = R @ old  :  (k x Kd) @ (Kd x N),  N = r_{j+1} * 4
    const int Kd = TL[j + 1];
    const int N  = TR[j + 1] * 4;
    lds_zero(rowbuf, 1024, lane);
    __syncthreads();
    for (int t = lane; t < Kd * N; t += NLANE) {
      int col = t % N; int rr = t / N;
      rowbuf[rr * 64 + col] = mps[(j + 1) * SLOT + rr * N + col];
    }
    __syncthreads();
    const int Kpad   = (Kd + 3) & ~3;
    const int Ntiles = (N + 15) >> 4;
    for (int nt = 0; nt < Ntiles; ++nt)
      mm16(P2, 16, rowbuf + nt * 16, 64, P0 + nt * 16, 64, Kpad, lane);
    __syncthreads();
    for (int t = lane; t < k * N; t += NLANE) {
      int col = t % N; int kk = t / N;
      mps[(j + 1) * SLOT + kk * N + col] = P0[kk * 64 + col];
    }
    if (lane == 0) { TR[j] = k; TL[j + 1] = k; }
    __syncthreads();
  }

  // ---- right-to-left truncated SVD sweep ----
  for (int j = LY - 1; j >= 1; --j) {
    const int l = TL[j], r = TR[j];
    const int cdim = r * 4;
    const int k = imin(CHI, imin(l, cdim));
    lds_zero(P0, 1024, lane);  // M^T, 64x16 ld16
    lds_zero(P1, 1024, lane);  // M,   16x64 ld64
    __syncthreads();
    for (int t = lane; t < l * cdim; t += NLANE) {
      int col = t % cdim; int row = t / cdim;
      float v = mps[j * SLOT + row * cdim + col];
      P1[row * 64 + col] = v;
      P0[col * 16 + row] = v;
    }
    __syncthreads();
    // Gram G = M M^T -> P2 (16x16 ld16)
    mm16(P1, 64, P0, 16, P2, 16, (cdim + 3) & ~3, lane);
    __syncthreads();
    // Jacobi eigensolver on G (n = l), eigenvectors in VE
    for (int t = lane; t < 256; t += NLANE) VE[t] = 0.f;
    __syncthreads();
    for (int t = lane; t < l; t += NLANE) VE[t * 16 + t] = 1.f;
    __syncthreads();
    for (int sweep = 0; sweep < 8; ++sweep) {
      for (int p = 0; p < l - 1; ++p)
        for (int q = p + 1; q < l; ++q) {
          float gpq = P2[p * 16 + q];
          float gpp = P2[p * 16 + p], gqq = P2[q * 16 + q];
          float cs = 1.f, sn = 0.f;
          if (fabsf(gpq) > 1e-12f * (fabsf(gpp) + fabsf(gqq)) + 1e-30f) {
            float tau = (gqq - gpp) / (2.f * gpq);
            float tt = (tau >= 0.f ? 1.f : -1.f) / (fabsf(tau) + sqrtf(1.f + tau * tau));
            cs = 1.f / sqrtf(1.f + tt * tt);
            sn = tt * cs;
          }
          __syncthreads();
          if (lane < l) {
            float gip = P2[lane * 16 + p], giq = P2[lane * 16 + q];
            P2[lane * 16 + p] = cs * gip - sn * giq;
            P2[lane * 16 + q] = sn * gip + cs * giq;
          }
          __syncthreads();
          if (lane < l) {
            float gpi = P2[p * 16 + lane], gqi = P2[q * 16 + lane];
            P2[p * 16 + lane] = cs * gpi - sn * gqi;
            P2[q * 16 + lane] = sn * gpi + cs * gqi;
          }
          __syncthreads();
          if (lane < l) {
            float vip = VE[lane * 16 + p], viq = VE[lane * 16 + q];
            VE[lane * 16 + p] = cs * vip - sn * viq;
            VE[lane * 16 + q] = sn * vip + cs * viq;
          }
          __syncthreads();
        }
    }
    // sort eigenvalues descending, take sqrt
    if (lane == 0) {
      for (int a = 0; a < l; ++a) { svals[a] = P2[a * 16 + a]; perm[a] = a; }
      for (int a = 0; a < l; ++a) {
        int best = a;
        for (int b = a + 1; b < l; ++b) if (svals[b] > svals[best]) best = b;
        float tv = svals[a]; svals[a] = svals[best]; svals[best] = tv;
        int tp = perm[a]; perm[a] = perm[best]; perm[best] = tp;
      }
      for (int a = 0; a < l; ++a) svals[a] = sqrtf(fmaxf(svals[a], 0.f));
    }
    __syncthreads();
    // Vh[:k] = diag(1/s) U^T M  ->  mps[j] (k, r, 4)
    lds_zero(P0, 256, lane);
    __syncthreads();
    for (int t = lane; t < k * l; t += NLANE) {
      int row = t % l; int i2 = t / l;
      P0[i2 * 16 + row] = VE[row * 16 + perm[i2]];
    }
    __syncthreads();
    const int Kp2 = (l + 3) & ~3;
    const int Nt2 = (cdim + 15) >> 4;
    for (int nt = 0; nt < Nt2; ++nt)
      mm16(P0, 16, P1 + nt * 16, 64, rowbuf + nt * 16, 64, Kp2, lane);
    __syncthreads();
    for (int t = lane; t < k * cdim; t += NLANE) {
      int col = t % cdim; int i2 = t / cdim;
      float s = svals[i2];
      float inv = (s > 1e-30f) ? 1.f / s : 0.f;
      mps[j * SLOT + i2 * cdim + col] = rowbuf[i2 * 64 + col] * inv;
    }
    __syncthreads();
    // mps[j-1] = einsum('lmu,mk->lku', old, U[:, :k]*S[:k])
    const int lo = TL[j - 1], m2 = TR[j - 1];  // m2 == l
    lds_zero(rowbuf, 1024, lane);  // A' = old permuted (l*u x m), 64x16 ld16
    lds_zero(P0, 256, lane);       // B  = U*S (m x k), 16x16 ld16
    __syncthreads();
    for (int t = lane; t < lo * 4 * m2; t += NLANE) {
      int mm = t % m2; int au = t / m2;
      int u = au & 3; int a = au >> 2;
      rowbuf[au * 16 + mm] = mps[(j - 1) * SLOT + (a * m2 + mm) * 4 + u];
    }
    for (int t = lane; t < m2 * k; t += NLANE) {
      int i2 = t % k; int mm = t / k;
      P0[mm * 16 + i2] = VE[mm * 16 + perm[i2]] * svals[i2];
    }
    __syncthreads();
    const int Kp3 = (m2 + 3) & ~3;
    const int Mt3 = (lo * 4 + 15) >> 4;
    for (int mt = 0; mt < Mt3; ++mt)
      mm16(rowbuf + mt * 256, 16, P0, 16, P2 + mt * 256, 16, Kp3, lane);
    __syncthreads();
    for (int t = lane; t < lo * k * 4; t += NLANE) {
      int u = t & 3; int rest = t >> 2;
      int i2 = rest % k; int a = rest / k;
      mps[(j - 1) * SLOT + (a * k + i2) * 4 + u] = P2[(a * 4 + u) * 16 + i2];
    }
    if (lane == 0) { TL[j] = k; TR[j - 1] = k; }
    __syncthreads();
  }

  // ---- strip tensor norms into logZ ----
  for (int j = 0; j < LY; ++j) {
    const int n = TL[j] * TR[j] * 4;
    float part = 0.f;
    for (int t = lane; t < n; t += NLANE) { float v = mps[j * SLOT + t]; part += v * v; }
    float nrm = sqrtf(wave_sum(part));
    float safe = (nrm > 0.f) ? nrm : 1.f;
    float inv = 1.f / safe;
    for (int t = lane; t < n; t += NLANE) mps[j * SLOT + t] *= inv;
    if (lane == 0) *logZ += logf(safe);
    __syncthreads();
  }
}

// Absorb one PEPS row into the boundary MPS (bottom: contract T.down; top: T.up)
static __device__ void absorb_row(float* mps, int* TL, int* TR,
                                  const float* peps, const int* x, int i, bool isTop,
                                  float* siteT, float* rowbuf, int lane) {
  for (int j = 0; j < LY; ++j) {
    int nu, nr2, nd, nl2;
    load_site(peps, x, i, j, siteT, nu, nr2, nd, nl2, lane);
    if (j + 1 < LY) {  // prefetch next site's slice
      int p2 = x[i * LY + (j + 1)];
      const float* nxt = peps + (((size_t)(i * LY + j + 1) * PPHYS + p2) << 8);
      __builtin_prefetch(nxt + lane * 8, 0, 1);
    }
    __syncthreads();
    const int bl = TL[j], br = TR[j];
    const int newL = bl * nl2, newR = br * nr2;
    const int total = newL * newR * 4;
    const int su = nr2 * nd * nl2, sr = nd * nl2, sd = nl2;
    for (int t = lane; t < total; t += NLANE) {
      int keep = t & 3; int rest = t >> 2;
      int col = rest % newR; int row = rest / newR;
      int ll = row / nl2, L = row % nl2;
      int rr = col / nr2, R = col % nr2;
      const float* Bp = &mps[j * SLOT + (ll * br + rr) * 4];
      float acc = 0.f;
      if (!isTop) {
        const float* Tp = &siteT[keep * su + R * sr + L];  // T[U][R][d][L], sum d
#pragma unroll
        for (int d = 0; d < 4; ++d) acc += Bp[d] * Tp[d * sd];
      } else {
        const float* Tp = &siteT[R * sr + keep * sd + L];  // T[t][R][D][L], sum t
#pragma unroll
        for (int d = 0; d < 4; ++d) acc += Bp[d] * Tp[d * su];
      }
      rowbuf[t] = acc;
    }
    __syncthreads();
    for (int t = lane; t < total; t += NLANE) mps[j * SLOT + t] = rowbuf[t];
    if (lane == 0) { TL[j] = newL; TR[j] = newR; }
    __syncthreads();
  }
}

__global__ __launch_bounds__(32, 1)
void peps_amp_kernel(const float* __restrict__ peps, const int* __restrict__ x,
                     float* __restrict__ out) {
  __shared__ float mps[LY * SLOT];
  __shared__ float rowbuf[SLOT];
  __shared__ float siteT[256];
  __shared__ float P0[SLOT];
  __shared__ float P1[SLOT];
  __shared__ float P2[SLOT];
  __shared__ float VE[256];
  __shared__ float botF[LY * 64];
  __shared__ float topF[LY * 64];
  __shared__ float vb[2][16];
  __shared__ float svals[16];
  __shared__ int perm[16];
  __shared__ int TL[LY], TR[LY];
  __shared__ int blF[LY], brF[LY], tlF[LY], trF[LY];
  __shared__ float logZ;
  const int lane = threadIdx.x;

  if (lane == 0) logZ = 0.f;
  __syncthreads();

  // ---------- bottom boundary: row 0 ----------
  for (int j = 0; j < LY; ++j) {
    int nu, nr2, nd, nl2;
    load_site(peps, x, 0, j, siteT, nu, nr2, nd, nl2, lane);
    __syncthreads();
    const int total = nl2 * nr2 * 4;  // (l, r, u)
    for (int t = lane; t < total; t += NLANE) {
      int u = t & 3; int rest = t >> 2;
      int r = rest % nr2; int l = rest / nr2;
      mps[j * SLOT + t] = siteT[(u * nr2 + r) * nl2 + l];  // nd == 1
    }
    if (lane == 0) { TL[j] = nl2; TR[j] = nr2; }
    __syncthreads();
  }
  for (int i = 1; i < LX / 2; ++i) {
    absorb_row(mps, TL, TR, peps, x, i, false, siteT, rowbuf, lane);
    compress_mps(mps, TL, TR, P0, P1, P2, rowbuf, VE, svals, perm, &logZ, lane);
  }
  for (int j = 0; j < LY; ++j) {
    const int n = TL[j] * TR[j] * 4;
    for (int t = lane; t < n; t += NLANE) botF[j * 64 + t] = mps[j * SLOT + t];
    if (lane == 0) { blF[j] = TL[j]; brF[j] = TR[j]; }
  }
  __syncthreads();

  // ---------- top boundary: row Lx-1 ----------
  for (int j = 0; j < LY; ++j) {
    int nu, nr2, nd, nl2;
    load_site(peps, x, LX - 1, j, siteT, nu, nr2, nd, nl2, lane);
    __syncthreads();
    const int total = nl2 * nr2 * 4;  // (l, r, d)
    for (int t = lane; t < total; t += NLANE) {
      int d = t & 3; int rest = t >> 2;
      int r = rest % nr2; int l = rest / nr2;
      mps[j * SLOT + t] = siteT[(r * nd + d) * nl2 + l];  // nu == 1
    }
    if (lane == 0) { TL[j] = nl2; TR[j] = nr2; }
    __syncthreads();
  }
  for (int i = LX - 2; i >= LX / 2; --i) {
    absorb_row(mps, TL, TR, peps, x, i, true, siteT, rowbuf, lane);
    compress_mps(mps, TL, TR, P0, P1, P2, rowbuf, VE, svals, perm, &logZ, lane);
  }
  for (int j = 0; j < LY; ++j) {
    const int n = TL[j] * TR[j] * 4;
    for (int t = lane; t < n; t += NLANE) topF[j * 64 + t] = mps[j * SLOT + t];
    if (lane == 0) { tlF[j] = TL[j]; trF[j] = TR[j]; }
  }
  __syncthreads();

  // ---------- final transfer-matrix contraction ----------
  if (lane == 0) vb[0][0] = 1.f;
  __syncthreads();
  int pa = 1, pb = 1, cur = 0;
  for (int j = 0; j < LY; ++j) {
    const int np = brF[j], nq = trF[j];
    const int tot = np * nq;
    for (int t = lane; t < tot; t += NLANE) {
      int q = t % nq, p = t / nq;
      float acc = 0.f;
      for (int a = 0; a < pa; ++a)
        for (int b = 0; b < pb; ++b) {
          float vab = vb[cur][a * pb + b];
#pragma unroll
          for (int u = 0; u < 4; ++u)
            acc += vab * botF[j * 64 + (a * np + p) * 4 + u] *
                          topF[j * 64 + (b * nq + q) * 4 + u];
        }
      vb[cur ^ 1][t] = acc;
    }
    cur ^= 1; pa = np; pb = nq;
    __syncthreads();
  }
  if (lane == 0) out[0] = vb[cur][0] * expf(logZ);
}

extern "C" void kernel_launch(void* const* d_in, const int* in_sizes, int n_in,
                              void* d_out, int out_size, void* d_ws, size_t ws_size,
                              hipStream_t stream) {
  (void)in_sizes; (void)n_in; (void)out_size; (void)d_ws; (void)ws_size;
  const float* peps = (const float*)d_in[0];
  const int* x = (const int*)d_in[1];
  float* out = (float*)d_out;
  peps_amp_kernel<<<1, 32, 0, stream>>>(peps, x, out);
}